// GravityFieldComputer_21869973471542
// MI455X (gfx1250) — compile-verified
//
#include <hip/hip_runtime.h>

typedef __attribute__((ext_vector_type(2))) float v2f;
typedef __attribute__((ext_vector_type(8))) float v8f;

#define GRAV_EPS2  1e-8f       // EPS^2: max(d2,EPS^2) == (max(d,EPS))^2
#define GN         4096        // N fixed by the reference
#define GTILES     256         // GN / 16
#define GTSHIFT    8           // log2(GTILES)

// Process one 16-wide j-tile against this wave's 16-row i-tile.
//   d2(16x16) = (-2*Xi)(16x4) * Xj(4x16) + [r2i + r2j]   via one V_WMMA_F32_16X16X4_F32
//   w = m_j*mask_j * rsq(d2)^3 ; (DIAG only) zero i==j ; 4 FMA accumulate per element
template <bool DIAG>
__device__ __forceinline__ void grav_tile(
    int jt, const float* __restrict__ posb, const float* __restrict__ masb,
    const float* __restrict__ mskb, bool lo, int col, v2f Am2,
    const float* __restrict__ r2i, const int* __restrict__ iglob,
    float* __restrict__ accX, float* __restrict__ accY,
    float* __restrict__ accZ, float* __restrict__ accW)
{
    const int jg = (jt << 4) + col;                  // batch-local j index
    const float* pj = posb + 3 * jg;
    const float xj = pj[0], yj = pj[1], zj = pj[2];
    const float mj  = masb[jg] * mskb[jg];           // m_j * mask_j
    const float r2j = fmaf(xj, xj, fmaf(yj, yj, zj * zj));

    // B (4x16): v0 = K0(x)/K2(z), v1 = K1(y)/K3(0)
    v2f Bm;
    Bm.x = lo ? xj : zj;
    Bm.y = lo ? yj : 0.0f;

    // C[m,n] = r2i[m] + r2j[n]  (both per-lane values in D layout)
    v8f C;
#pragma unroll
    for (int r = 0; r < 8; ++r) C[r] = r2i[r] + r2j;

    // D = (-2*Xi)*Xj + C = squared distances, directly
    const v8f d2v = __builtin_amdgcn_wmma_f32_16x16x4_f32(
        false, Am2, false, Bm, (short)0, C, false, false);

#pragma unroll
    for (int r = 0; r < 8; ++r) {
        const float d2   = fmaxf(d2v[r], GRAV_EPS2);
        const float inv  = __builtin_amdgcn_rsqf(d2);     // v_rsq_f32 (trans pipe)
        float w = (mj * inv) * (inv * inv);               // m_j / d^3
        if (DIAG) w = (jg == iglob[r]) ? 0.0f : w;        // zero self-pair
        accX[r] = fmaf(w, xj, accX[r]);
        accY[r] = fmaf(w, yj, accY[r]);
        accZ[r] = fmaf(w, zj, accZ[r]);
        accW[r] += w;
    }
}

// grid = (totalWaves/4, S): blockIdx.y selects a j-segment; each wave owns one
// 16-row i-tile of one batch and writes 16 force rows into its segment's slab.
__global__ __launch_bounds__(128) void grav_wmma_kernel(
    const float* __restrict__ pos,   // (B,N,3)
    const float* __restrict__ mas,   // (B,N,1)
    const float* __restrict__ Gp,    // (1,)
    const float* __restrict__ msk,   // (B,N)
    float* __restrict__ dst,         // S slabs of (B,N,3)
    int tilesPerSeg, int outElems)
{
    const int lane  = threadIdx.x & 31;
    const int wave  = threadIdx.x >> 5;
    const int gwave = blockIdx.x * 4 + wave;
    const int b     = gwave >> GTSHIFT;
    const int tile  = gwave & (GTILES - 1);
    const int iBase = tile << 4;

    const int seg = blockIdx.y;
    const int jt0 = seg * tilesPerSeg;
    const int jt1 = jt0 + tilesPerSeg;
    dst += (size_t)seg * outElems;

    const int  col = lane & 15;
    const bool lo  = lane < 16;

    const float* posb = pos + (size_t)b * GN * 3;
    const float* masb = mas + (size_t)b * GN;
    const float* mskb = msk + (size_t)b * GN;
    float*       outb = dst + (size_t)b * GN * 3;

    const float Gv = Gp[0];

    // A pre-scaled by -2 (16x4): v0 = K0(-2x)/K2(-2z), v1 = K1(-2y)/K3(0).
    // Load all three coords unconditionally (no EXEC-predicated load), then select.
    const int iLane = iBase + col;
    const float* pi = posb + 3 * iLane;
    const float axL = pi[0], ayL = pi[1], azL = pi[2];
    v2f Am2;
    Am2.x = -2.0f * (lo ? axL : azL);
    Am2.y = lo ? (-2.0f * ayL) : 0.0f;

    // Per-row i data: m = r (lanes 0-15) / r+8 (lanes 16-31)
    float xi[8], yi[8], zi[8], r2i[8], gm[8];
    int   iglob[8];
    float accX[8], accY[8], accZ[8], accW[8];
#pragma unroll
    for (int r = 0; r < 8; ++r) {
        const int ig = iBase + r + (lo ? 0 : 8);
        const float x = posb[3 * ig + 0];
        const float y = posb[3 * ig + 1];
        const float z = posb[3 * ig + 2];
        xi[r] = x; yi[r] = y; zi[r] = z;
        r2i[r]   = fmaf(x, x, fmaf(y, y, z * z));
        gm[r]    = masb[ig] * mskb[ig];       // m_i * mask_i
        iglob[r] = ig;
        accX[r] = 0.0f; accY[r] = 0.0f; accZ[r] = 0.0f; accW[r] = 0.0f;
    }

    // Split the j-range around the (wave-uniform) self-tile so the hot loops
    // are branch-free inside and unrollable.
    const bool hasDiag = (tile >= jt0 && tile < jt1);
    const int  endA    = hasDiag ? tile : jt1;
    const int  startB  = hasDiag ? tile + 1 : jt1;

#pragma unroll 2
    for (int jt = jt0; jt < endA; ++jt)
        grav_tile<false>(jt, posb, masb, mskb, lo, col, Am2,
                         r2i, iglob, accX, accY, accZ, accW);
#pragma unroll 2
    for (int jt = startB; jt < jt1; ++jt)
        grav_tile<false>(jt, posb, masb, mskb, lo, col, Am2,
                         r2i, iglob, accX, accY, accZ, accW);
    if (hasDiag)
        grav_tile<true>(tile, posb, masb, mskb, lo, col, Am2,
                        r2i, iglob, accX, accY, accZ, accW);

    // Reduce 16 column-partials within each half-wave; lanes 0/16 write rows
#pragma unroll
    for (int r = 0; r < 8; ++r) {
#pragma unroll
        for (int m = 1; m <= 8; m <<= 1) {
            accX[r] += __shfl_xor(accX[r], m, 32);
            accY[r] += __shfl_xor(accY[r], m, 32);
            accZ[r] += __shfl_xor(accZ[r], m, 32);
            accW[r] += __shfl_xor(accW[r], m, 32);
        }
        if (col == 0) {
            const int   i = iglob[r];
            const float s = Gv * gm[r];       // G * m_i * mask_i
            outb[3 * i + 0] = s * fmaf(-xi[r], accW[r], accX[r]);
            outb[3 * i + 1] = s * fmaf(-yi[r], accW[r], accY[r]);
            outb[3 * i + 2] = s * fmaf(-zi[r], accW[r], accZ[r]);
        }
    }
}

// Sum S partial-force slabs (force is linear in the accumulators).
__global__ void grav_reduce_kernel(const float* __restrict__ ws,
                                   float* __restrict__ out, int n, int S)
{
    const int idx = blockIdx.x * 256 + threadIdx.x;
    if (idx < n) {
        float s = 0.0f;
        for (int k = 0; k < S; ++k) s += ws[(size_t)k * n + idx];
        out[idx] = s;
    }
}

extern "C" void kernel_launch(void* const* d_in, const int* in_sizes, int n_in,
                              void* d_out, int out_size, void* d_ws, size_t ws_size,
                              hipStream_t stream) {
    const float* pos = (const float*)d_in[0];
    const float* mas = (const float*)d_in[1];
    const float* Gp  = (const float*)d_in[2];
    const float* msk = (const float*)d_in[3];
    float*       out = (float*)d_out;

    const int BN = in_sizes[3];               // B*N from mask
    const int B  = BN / GN;
    const int totalWaves = B * GTILES;        // full wave32s, EXEC all-1s
    const int wavesPerBlock = 4;
    const int outElems = B * GN * 3;

    // 4-way j-split for occupancy if the workspace can hold the slabs.
    int S = 4;
    if ((size_t)S * (size_t)outElems * sizeof(float) > ws_size) S = 1;

    const int tilesPerSeg = GTILES / S;
    float* dst = (S > 1) ? (float*)d_ws : out;

    dim3 grid(totalWaves / wavesPerBlock, S);
    dim3 block(wavesPerBlock * 32);
    grav_wmma_kernel<<<grid, block, 0, stream>>>(pos, mas, Gp, msk, dst,
                                                 tilesPerSeg, outElems);
    if (S > 1) {
        dim3 rgrid((outElems + 255) / 256);
        grav_reduce_kernel<<<rgrid, dim3(256), 0, stream>>>((const float*)d_ws,
                                                            out, outElems, S);
    }
}